// SuperPointMatching_65292092833933
// MI455X (gfx1250) — compile-verified
//
#include <hip/hip_runtime.h>
#include <hip/hip_bf16.h>

typedef unsigned int uint_t;
typedef float v2f __attribute__((ext_vector_type(2)));
typedef float v8f __attribute__((ext_vector_type(8)));

#define NREF 5000
#define NSRC 5000
#define KF   256
#define TM   64
#define TN   64
#define KC   64
#define TILES 79            // ceil(5000/64) -> 79*64 = 5056
#define NPAD  (TILES * 64)  // 5056
#define NCORR 4096

#define LDS_STRIDE (KC + 2)          // 66 floats, kills bank conflicts
#define LDS_A_OFF  0                 // float index of As within shmem
#define LDS_B_OFF  (TM * LDS_STRIDE) // 4224
#define LDS_E_OFF  (2 * TM * LDS_STRIDE)  // 8448
#define LDS_TOTAL  (3 * TM * LDS_STRIDE)  // 12672 floats = 50688 B

// ---------------------------------------------------------------------------
// fp32 WMMA wrapper: D = A(16x4) * B(4x16) + C  (exact fp32, CDNA5 matrix pipe)
// ---------------------------------------------------------------------------
__device__ __forceinline__ v8f wmma4(v2f a, v2f b, v8f c) {
  return __builtin_amdgcn_wmma_f32_16x16x4_f32(false, a, false, b, (short)0, c,
                                               false, false);
}

// ---------------------------------------------------------------------------
// threefry2x32 (JAX PRNG) for the FPS initial index
// ---------------------------------------------------------------------------
__device__ __forceinline__ uint_t rotl32(uint_t x, int r) {
  return (x << r) | (x >> (32 - r));
}
__device__ uint_t threefry2x32(uint_t k0, uint_t k1, uint_t c0, uint_t c1) {
  uint_t ks2 = 0x1BD11BDAu ^ k0 ^ k1;
  uint_t x0 = c0 + k0, x1 = c1 + k1;
  const int ra0 = 13, ra1 = 15, ra2 = 26, ra3 = 6;
  const int rb0 = 17, rb1 = 29, rb2 = 16, rb3 = 24;
#define TF_R(r) { x0 += x1; x1 = rotl32(x1, r); x1 ^= x0; }
  TF_R(ra0) TF_R(ra1) TF_R(ra2) TF_R(ra3) x0 += k1;  x1 += ks2 + 1u;
  TF_R(rb0) TF_R(rb1) TF_R(rb2) TF_R(rb3) x0 += ks2; x1 += k0 + 2u;
  TF_R(ra0) TF_R(ra1) TF_R(ra2) TF_R(ra3) x0 += k0;  x1 += k1 + 3u;
  TF_R(rb0) TF_R(rb1) TF_R(rb2) TF_R(rb3) x0 += k1;  x1 += ks2 + 4u;
  TF_R(ra0) TF_R(ra1) TF_R(ra2) TF_R(ra3) x0 += ks2; x1 += k0 + 5u;
#undef TF_R
  return x0;
}

// ---------------------------------------------------------------------------
// zero a uint region
// ---------------------------------------------------------------------------
__global__ void initZeroKernel(uint_t* p, int n) {
  int i = blockIdx.x * blockDim.x + threadIdx.x;
  if (i < n) p[i] = 0u;
}

// ---------------------------------------------------------------------------
// Fused GEMM pass. MODE: 0=row/col partial sums, 1..3=histogram levels,
// 4=candidate collection. Recomputes scores bit-identically each pass.
// Block = 256 threads (8 wave32). 64x64 tile, K chunked by 64 through LDS.
// Interior tiles stream A/B with GLOBAL_LOAD_ASYNC_TO_LDS_B128 (ASYNCcnt);
// boundary tiles use the masked register path (zero padding).
// ---------------------------------------------------------------------------
template <int MODE>
__global__ __launch_bounds__(256) void scoresPass(
    const float* __restrict__ ref, const float* __restrict__ src,
    float* __restrict__ rowPart, float* __restrict__ colPart,
    const float* __restrict__ rowsum, const float* __restrict__ colsum,
    uint_t* __restrict__ hist, uint_t* __restrict__ ctrl,
    float* __restrict__ candS, uint_t* __restrict__ candI,
    uint_t* __restrict__ ties) {
  // Single shared allocation => known LDS byte offsets (base 0) for the
  // async-to-LDS VDST operands.
  __shared__ __align__(16) float shmem[LDS_TOTAL];
  float (*As)[LDS_STRIDE] = (float (*)[LDS_STRIDE])(shmem + LDS_A_OFF);
  float (*Bs)[LDS_STRIDE] = (float (*)[LDS_STRIDE])(shmem + LDS_B_OFF);
  float* Ex = shmem + LDS_E_OFF;  // mode0: E-tile[64][65]; else rs/cs + hist

  const int tid  = threadIdx.x;
  const int lane = tid & 31;
  const int wid  = tid >> 5;
  const int sr   = wid >> 2;   // 0..1
  const int sc   = wid & 3;    // 0..3
  const int tileR = blockIdx.y * TM;
  const int tileC = blockIdx.x * TN;
  const bool interior = (blockIdx.x < TILES - 1) && (blockIdx.y < TILES - 1);

  float* shRS = Ex;        // [64] row sums for this tile's rows
  float* shCS = Ex + 64;   // [64] col sums for this tile's cols
  uint_t* lh  = (uint_t*)(Ex + 128);  // local histogram (<=4096 bins)

  if (MODE >= 1) {
    if (tid < 64)  shRS[tid] = rowsum[tileR + tid];
    else if (tid < 128) shCS[tid - 64] = colsum[tileC + (tid - 64)];
    if (MODE <= 3) {
      const int nb = (MODE == 3) ? 256 : 4096;
      for (int i = tid; i < nb; i += 256) lh[i] = 0u;
    }
  }

  v8f acc0 = {};
  v8f acc1 = {};

  const int arow = lane & 15;
  const int koff = (lane >> 4) << 1;
  const int r0 = sr * 16 + arow;
  const int r1 = (sr + 2) * 16 + arow;
  const int bc = sc * 16 + (lane & 15);

  for (int kc = 0; kc < KF; kc += KC) {
    if (interior) {
      // async DMA: memory -> LDS, no VGPR round trip, tracked by ASYNCcnt
      for (int v = tid; v < TM * (KC / 4); v += 256) {
        int r  = v >> 4;
        int c4 = (v & 15) << 2;
        unsigned ldsA  = (unsigned)((LDS_A_OFF + r * LDS_STRIDE + c4) * 4);
        unsigned ldsB  = (unsigned)((LDS_B_OFF + r * LDS_STRIDE + c4) * 4);
        unsigned gOffA = (unsigned)(((tileR + r) * KF + kc + c4) * 4);
        unsigned gOffB = (unsigned)(((tileC + r) * KF + kc + c4) * 4);
        asm volatile("global_load_async_to_lds_b128 %0, %1, %2"
                     :: "v"(ldsA), "v"(gOffA), "s"(ref) : "memory");
        asm volatile("global_load_async_to_lds_b128 %0, %1, %2"
                     :: "v"(ldsB), "v"(gOffB), "s"(src) : "memory");
      }
      asm volatile("s_wait_asynccnt 0x0" ::: "memory");
    } else {
      // boundary tile: masked loads with zero padding
      for (int v = tid; v < TM * (KC / 4); v += 256) {
        int r  = v >> 4;
        int c4 = (v & 15) << 2;
        int gr = tileR + r;
        float4 val;
        if (gr < NREF) {
          val = *reinterpret_cast<const float4*>(ref + (size_t)gr * KF + kc + c4);
          if (kc + KC < KF)
            __builtin_prefetch(ref + (size_t)gr * KF + kc + KC + c4, 0, 1);
        } else {
          val = make_float4(0.f, 0.f, 0.f, 0.f);
        }
        As[r][c4 + 0] = val.x; As[r][c4 + 1] = val.y;
        As[r][c4 + 2] = val.z; As[r][c4 + 3] = val.w;

        int gc = tileC + r;
        float4 bv;
        if (gc < NSRC) {
          bv = *reinterpret_cast<const float4*>(src + (size_t)gc * KF + kc + c4);
          if (kc + KC < KF)
            __builtin_prefetch(src + (size_t)gc * KF + kc + KC + c4, 0, 1);
        } else {
          bv = make_float4(0.f, 0.f, 0.f, 0.f);
        }
        Bs[r][c4 + 0] = bv.x; Bs[r][c4 + 1] = bv.y;
        Bs[r][c4 + 2] = bv.z; Bs[r][c4 + 3] = bv.w;
      }
    }
    __syncthreads();

#pragma unroll
    for (int k = 0; k < KC; k += 4) {
      v2f a0 = *reinterpret_cast<const v2f*>(&As[r0][k + koff]);
      v2f a1 = *reinterpret_cast<const v2f*>(&As[r1][k + koff]);
      v2f bb = *reinterpret_cast<const v2f*>(&Bs[bc][k + koff]);
      acc0 = wmma4(a0, bb, acc0);
      acc1 = wmma4(a1, bb, acc1);
    }
    __syncthreads();
  }

  const int n  = lane & 15;
  const int hi = lane >> 4;

  if (MODE == 0) {
    // e = exp(2*dot - 2), masked; tile-local row/col partial sums (no atomics)
    float* Et = Ex;  // [64][65]
#pragma unroll
    for (int half = 0; half < 2; ++half) {
      const int rowBase = (sr + 2 * half) * 16;
#pragma unroll
      for (int r = 0; r < 8; ++r) {
        int m  = rowBase + r + 8 * hi;
        int gi = tileR + m, gj = tileC + sc * 16 + n;
        float dot = half ? acc1[r] : acc0[r];
        float e = (gi < NREF && gj < NSRC) ? expf(2.0f * dot - 2.0f) : 0.0f;
        Et[m * 65 + sc * 16 + n] = e;
      }
    }
    __syncthreads();
    if (tid < 64) {
      float s = 0.f;
      for (int c = 0; c < 64; ++c) s += Et[tid * 65 + c];
      rowPart[(size_t)(tileR + tid) * TILES + blockIdx.x] = s;
    } else if (tid < 128) {
      int c = tid - 64;
      float s = 0.f;
      for (int r = 0; r < 64; ++r) s += Et[r * 65 + c];
      colPart[(size_t)(tileC + c) * TILES + blockIdx.y] = s;
    }
  } else {
    uint_t b1v = 0, b2v = 0, Tv = 0;
    if (MODE == 2) b1v = ctrl[0];
    if (MODE == 3) { b1v = ctrl[0]; b2v = ctrl[2]; }
    if (MODE == 4) { Tv = ctrl[4]; }
#pragma unroll
    for (int half = 0; half < 2; ++half) {
      const int rowBase = (sr + 2 * half) * 16;
#pragma unroll
      for (int r = 0; r < 8; ++r) {
        int m  = rowBase + r + 8 * hi;
        int gi = tileR + m, gj = tileC + sc * 16 + n;
        if (gi < NREF && gj < NSRC) {
          float dot = half ? acc1[r] : acc0[r];
          float e = expf(2.0f * dot - 2.0f);
          float f = (e * e) / (shRS[m] * shCS[sc * 16 + n]);
          uint_t u = __float_as_uint(f);
          if (MODE == 1) {
            atomicAdd(&lh[u >> 20], 1u);
          } else if (MODE == 2) {
            if ((u >> 20) == b1v) atomicAdd(&lh[(u >> 8) & 0xFFFu], 1u);
          } else if (MODE == 3) {
            if ((u >> 20) == b1v && ((u >> 8) & 0xFFFu) == b2v)
              atomicAdd(&lh[u & 0xFFu], 1u);
          } else {  // MODE == 4: collect
            uint_t flat = (uint_t)gi * (uint_t)NSRC + (uint_t)gj;
            if (u > Tv) {
              uint_t p = atomicAdd(&ctrl[7], 1u);
              if (p < (uint_t)NCORR) { candS[p] = f; candI[p] = flat; }
            } else if (u == Tv) {
              uint_t p = atomicAdd(&ctrl[8], 1u);
              if (p < (uint_t)NCORR) ties[p] = flat;
            }
          }
        }
      }
    }
    if (MODE <= 3) {
      __syncthreads();
      const int nb = (MODE == 3) ? 256 : 4096;
      for (int i = tid; i < nb; i += 256)
        if (lh[i]) atomicAdd(&hist[i], lh[i]);
    }
  }
}

// ---------------------------------------------------------------------------
// fixed-order reduction of tile partials (deterministic, no float atomics)
// ---------------------------------------------------------------------------
__global__ void reduceSumsKernel(const float* __restrict__ rowPart,
                                 const float* __restrict__ colPart,
                                 float* __restrict__ rowsum,
                                 float* __restrict__ colsum) {
  int i = blockIdx.x * blockDim.x + threadIdx.x;
  if (i < NPAD) {
    float s = 0.f;
    for (int t = 0; t < TILES; ++t) s += rowPart[(size_t)i * TILES + t];
    rowsum[i] = s;
  } else if (i < 2 * NPAD) {
    int c = i - NPAD;
    float s = 0.f;
    for (int t = 0; t < TILES; ++t) s += colPart[(size_t)c * TILES + t];
    colsum[c] = s;
  }
}

// ---------------------------------------------------------------------------
// top-down histogram scan: finds the radix bucket holding the 4096th element
// ---------------------------------------------------------------------------
__global__ void scanHistKernel(const uint_t* __restrict__ h, int nbins,
                               uint_t* __restrict__ ctrl, int level) {
  if (threadIdx.x != 0 || blockIdx.x != 0) return;
  uint_t prev = 0;
  if (level == 2) prev = ctrl[1];
  if (level == 3) prev = ctrl[3];
  uint_t target = (uint_t)NCORR - prev;
  uint_t cum = 0;
  int b = 0;
  for (int i = nbins - 1; i >= 0; --i) {
    uint_t c = h[i];
    if (cum + c >= target) { b = i; break; }
    cum += c;
  }
  if (level == 1) {
    ctrl[0] = (uint_t)b; ctrl[1] = cum;
  } else if (level == 2) {
    ctrl[2] = (uint_t)b; ctrl[3] = prev + cum;
  } else {
    uint_t cAbove = prev + cum;
    uint_t T = (ctrl[0] << 20) | (ctrl[2] << 8) | (uint_t)b;
    ctrl[4] = T;
    ctrl[5] = cAbove;
    ctrl[6] = (uint_t)NCORR - cAbove;
  }
}

// ---------------------------------------------------------------------------
// resolve exact-score ties: lowest flat index wins (lax.top_k order)
// ---------------------------------------------------------------------------
__global__ __launch_bounds__(512) void tieResolveKernel(
    const uint_t* __restrict__ ties, const uint_t* __restrict__ ctrl,
    float* __restrict__ candS, uint_t* __restrict__ candI) {
  __shared__ uint_t tv[NCORR];
  int tid = threadIdx.x;
  uint_t cnt = ctrl[8];
  if (cnt > (uint_t)NCORR) cnt = (uint_t)NCORR;
  uint_t T = ctrl[4], cAbove = ctrl[5], need = ctrl[6];
  for (int i = tid; i < NCORR; i += 512)
    tv[i] = ((uint_t)i < cnt) ? ties[i] : 0xFFFFFFFFu;
  __syncthreads();
  for (int k = 2; k <= NCORR; k <<= 1) {
    for (int j = k >> 1; j > 0; j >>= 1) {
      for (int i = tid; i < NCORR; i += 512) {
        int ixj = i ^ j;
        if (ixj > i) {
          uint_t a = tv[i], b = tv[ixj];
          bool asc = ((i & k) == 0);
          if (asc ? (a > b) : (a < b)) { tv[i] = b; tv[ixj] = a; }
        }
      }
      __syncthreads();
    }
  }
  float Tf = __uint_as_float(T);
  for (int i = tid; i < (int)need; i += 512) {
    candS[cAbove + i] = Tf;
    candI[cAbove + i] = tv[i];
  }
}

// ---------------------------------------------------------------------------
// in-LDS bitonic sort of 4096 (score desc, idx asc) pairs -> top_k ordering
// ---------------------------------------------------------------------------
__global__ __launch_bounds__(512) void sortCandKernel(float* __restrict__ cs_,
                                                      uint_t* __restrict__ ci_) {
  __shared__ float ks[NCORR];
  __shared__ uint_t vs[NCORR];
  int tid = threadIdx.x;
  for (int i = tid; i < NCORR; i += 512) { ks[i] = cs_[i]; vs[i] = ci_[i]; }
  __syncthreads();
  for (int k = 2; k <= NCORR; k <<= 1) {
    for (int j = k >> 1; j > 0; j >>= 1) {
      for (int i = tid; i < NCORR; i += 512) {
        int ixj = i ^ j;
        if (ixj > i) {
          float s0 = ks[i], s1 = ks[ixj];
          uint_t v0 = vs[i], v1 = vs[ixj];
          bool greater = (s0 > s1) || (s0 == s1 && v0 < v1);
          bool desc = ((i & k) == 0);
          if (desc ? !greater : greater) {
            ks[i] = s1; ks[ixj] = s0; vs[i] = v1; vs[ixj] = v0;
          }
        }
      }
      __syncthreads();
    }
  }
  for (int i = tid; i < NCORR; i += 512) { cs_[i] = ks[i]; ci_[i] = vs[i]; }
}

// ---------------------------------------------------------------------------
// farthest point sampling: both runs fused into one launch (2 blocks on
// separate WGPs run concurrently). Points LDS-resident, 5000 serial
// iterations, wave32 shuffle argmax + 2 barriers per iteration.
// ---------------------------------------------------------------------------
__global__ __launch_bounds__(256) void fpsKernel(
    const float* __restrict__ refPts, const float* __restrict__ srcPts,
    const uint_t* __restrict__ candI, int* __restrict__ fpsRef,
    int* __restrict__ fpsSrc) {
  __shared__ __align__(16) float ptsL[NCORR * 3];
  __shared__ float wMax[8];
  __shared__ int wArg[8];
  __shared__ int sFar;

  const int useRef = (blockIdx.x == 0);
  const float* pts = useRef ? refPts : srcPts;
  int* outIdx = useRef ? fpsRef : fpsSrc;
  const uint_t seed = useRef ? 1u : 2u;

  const int tid = threadIdx.x, lane = tid & 31, wid = tid >> 5;
  const int PT = NCORR / 256;  // 16 points per thread
  const int base = tid * PT;

  float px[16], py[16], pz[16], dmin[16];
#pragma unroll
  for (int s = 0; s < PT; ++s) {
    uint_t flat = candI[base + s];
    uint_t pi = useRef ? (flat / (uint_t)NSRC) : (flat % (uint_t)NSRC);
    if (pi >= (uint_t)NSRC) pi = (uint_t)NSRC - 1u;
    float x = pts[pi * 3 + 0], y = pts[pi * 3 + 1], z = pts[pi * 3 + 2];
    px[s] = x; py[s] = y; pz[s] = z; dmin[s] = 1e10f;
    ptsL[(base + s) * 3 + 0] = x;
    ptsL[(base + s) * 3 + 1] = y;
    ptsL[(base + s) * 3 + 2] = z;
  }
  if (tid == 0) sFar = (int)(threefry2x32(0u, seed, 0u, 0u) % (uint_t)NCORR);
  __syncthreads();

  for (int it = 0; it < NREF; ++it) {
    int far = sFar;
    if (tid == 0) outIdx[it] = far;
    float cx = ptsL[far * 3 + 0], cy = ptsL[far * 3 + 1], cz = ptsL[far * 3 + 2];
    float best = -1.0f;
    int bi = base;
#pragma unroll
    for (int s = 0; s < PT; ++s) {
      float dx = px[s] - cx, dy = py[s] - cy, dz = pz[s] - cz;
      float d = dx * dx + dy * dy + dz * dz;
      d = fminf(dmin[s], d);
      dmin[s] = d;
      if (d > best) { best = d; bi = base + s; }
    }
    // wave32 argmax (lower index wins ties)
    for (int off = 16; off > 0; off >>= 1) {
      float ov = __shfl_down(best, off, 32);
      int oi = __shfl_down(bi, off, 32);
      if (ov > best || (ov == best && oi < bi)) { best = ov; bi = oi; }
    }
    if (lane == 0) { wMax[wid] = best; wArg[wid] = bi; }
    __syncthreads();
    if (tid == 0) {
      float b = wMax[0];
      int a = wArg[0];
      for (int w = 1; w < 8; ++w)
        if (wMax[w] > b || (wMax[w] == b && wArg[w] < a)) { b = wMax[w]; a = wArg[w]; }
      sFar = a;
    }
    __syncthreads();
  }
}

// ---------------------------------------------------------------------------
// final epilogue: drop-zero/append-0 compaction, pick longer FPS list,
// gather indices + recompute the 256 output scores
// ---------------------------------------------------------------------------
__global__ __launch_bounds__(256) void assembleKernel(
    const int* __restrict__ fpsRef, const int* __restrict__ fpsSrc,
    const uint_t* __restrict__ candI, const float* __restrict__ ref,
    const float* __restrict__ src, const float* __restrict__ rowsum,
    const float* __restrict__ colsum, float* __restrict__ out) {
  __shared__ int fsel[256];
  if (threadIdx.x == 0) {
    int cfr = 1, cfs = 1;  // the appended trailing 0
    for (int i = 0; i < NREF; ++i) cfr += (fpsRef[i] != 0);
    for (int i = 0; i < NSRC; ++i) cfs += (fpsSrc[i] != 0);
    const int* chosen = (cfr > cfs) ? fpsRef : fpsSrc;
    int cn = (cfr > cfs) ? NREF : NSRC;
    int j = 0;
    for (int i = 0; i < cn && j < 256; ++i)
      if (chosen[i] != 0) fsel[j++] = chosen[i];
    while (j < 256) fsel[j++] = 0;
  }
  __syncthreads();
  int t = threadIdx.x;
  int ci = fsel[t];
  if (ci < 0 || ci >= NCORR) ci = 0;
  uint_t flat = candI[ci];
  uint_t r = flat / (uint_t)NSRC;
  uint_t c = flat % (uint_t)NSRC;
  if (r >= (uint_t)NREF) r = NREF - 1;
  float dot = 0.f;
  for (int k = 0; k < KF; ++k)
    dot += ref[(size_t)r * KF + k] * src[(size_t)c * KF + k];
  float e = expf(2.0f * dot - 2.0f);
  float f = (e * e) / (rowsum[r] * colsum[c]);
  out[t]       = (float)r;
  out[256 + t] = (float)c;
  out[512 + t] = f;
}

// ---------------------------------------------------------------------------
extern "C" void kernel_launch(void* const* d_in, const int* in_sizes, int n_in,
                              void* d_out, int out_size, void* d_ws,
                              size_t ws_size, hipStream_t stream) {
  (void)in_sizes; (void)n_in; (void)out_size; (void)ws_size;
  const float* ref_feats  = (const float*)d_in[0];
  const float* src_feats  = (const float*)d_in[1];
  const float* ref_points = (const float*)d_in[2];
  const float* src_points = (const float*)d_in[3];
  float* out = (float*)d_out;

  // workspace layout
  float* rowsum  = (float*)d_ws;                       // NPAD
  float* colsum  = rowsum + NPAD;                      // NPAD
  float* rowPart = colsum + NPAD;                      // NPAD*TILES
  float* colPart = rowPart + (size_t)NPAD * TILES;     // NPAD*TILES
  uint_t* h1     = (uint_t*)(colPart + (size_t)NPAD * TILES);  // 4096
  uint_t* h2     = h1 + 4096;                          // 4096
  uint_t* h3     = h2 + 4096;                          // 256
  uint_t* ctrl   = h3 + 256;                           // 64
  float* candS   = (float*)(ctrl + 64);                // 4096
  uint_t* candI  = (uint_t*)(candS + NCORR);           // 4096
  uint_t* ties   = candI + NCORR;                      // 4096
  int* fpsRef    = (int*)(ties + NCORR);               // NPAD
  int* fpsSrc    = fpsRef + NPAD;                      // NPAD

  const int zeroCount = 4096 + 4096 + 256 + 64 + NCORR + NCORR + NCORR;
  initZeroKernel<<<(zeroCount + 255) / 256, 256, 0, stream>>>(h1, zeroCount);

  dim3 tgrid(TILES, TILES);
  // pass 0: row/col sums of e = exp(2*dot - 2)
  scoresPass<0><<<tgrid, 256, 0, stream>>>(ref_feats, src_feats, rowPart,
      colPart, rowsum, colsum, h1, ctrl, candS, candI, ties);
  reduceSumsKernel<<<(2 * NPAD + 255) / 256, 256, 0, stream>>>(rowPart, colPart,
                                                               rowsum, colsum);
  // 3-level radix histogram on float bits of final score
  scoresPass<1><<<tgrid, 256, 0, stream>>>(ref_feats, src_feats, rowPart,
      colPart, rowsum, colsum, h1, ctrl, candS, candI, ties);
  scanHistKernel<<<1, 1, 0, stream>>>(h1, 4096, ctrl, 1);
  scoresPass<2><<<tgrid, 256, 0, stream>>>(ref_feats, src_feats, rowPart,
      colPart, rowsum, colsum, h2, ctrl, candS, candI, ties);
  scanHistKernel<<<1, 1, 0, stream>>>(h2, 4096, ctrl, 2);
  scoresPass<3><<<tgrid, 256, 0, stream>>>(ref_feats, src_feats, rowPart,
      colPart, rowsum, colsum, h3, ctrl, candS, candI, ties);
  scanHistKernel<<<1, 1, 0, stream>>>(h3, 256, ctrl, 3);
  // collect: everything strictly above exact threshold + tie pool
  scoresPass<4><<<tgrid, 256, 0, stream>>>(ref_feats, src_feats, rowPart,
      colPart, rowsum, colsum, h1, ctrl, candS, candI, ties);
  tieResolveKernel<<<1, 512, 0, stream>>>(ties, ctrl, candS, candI);
  sortCandKernel<<<1, 512, 0, stream>>>(candS, candI);

  // both FPS runs concurrently (2 blocks -> 2 WGPs)
  fpsKernel<<<2, 256, 0, stream>>>(ref_points, src_points, candI, fpsRef,
                                   fpsSrc);

  assembleKernel<<<1, 256, 0, stream>>>(fpsRef, fpsSrc, candI, ref_feats,
                                        src_feats, rowsum, colsum, out);
}